// GELU64_17566416240640
// MI455X (gfx1250) — compile-verified
//
#include <hip/hip_runtime.h>
#include <math.h>

// Problem geometry (fixed by the reference).
#define B_LEN 8
#define T_LEN 4096
#define D_LEN 1024
#define DT    32                 // channels per workgroup (== wave32)
#define NBLK  (D_LEN / DT)       // 32 d-blocks -> 256 workgroups
#define CH    32                 // timesteps per staged chunk (4 KB tile)
#define NCHUNK (T_LEN / CH)      // 128

// ---- CDNA5 feature probes (device pass only; host pass parses fallbacks) ----
#if defined(__HIP_DEVICE_COMPILE__) && __has_builtin(__builtin_amdgcn_global_load_async_to_lds_b128)
#define HAVE_ASYNC128 1
#else
#define HAVE_ASYNC128 0
#endif
#if defined(__HIP_DEVICE_COMPILE__) && __has_builtin(__builtin_amdgcn_global_load_async_to_lds_b32)
#define HAVE_ASYNC32 1
#else
#define HAVE_ASYNC32 0
#endif
#define HAVE_ASYNC (HAVE_ASYNC128 || HAVE_ASYNC32)

#if HAVE_ASYNC
typedef __attribute__((address_space(1))) int GInt;   // global ('__device__ int*')
typedef __attribute__((address_space(3))) int LInt;   // LDS
typedef int v4i __attribute__((vector_size(16)));     // 4 x int (16 B)
typedef __attribute__((address_space(1))) v4i GV4;    // global v4i
typedef __attribute__((address_space(3))) v4i LV4;    // LDS v4i
#endif

#if HAVE_ASYNC128
#define NBUF   7                 // 7 x 4KB tile buffers
#define LOOKA  6                 // chunks in flight beyond current
#define WAIT_LANDED 48           // 6 chunks x 8 b128-instructions
#else
#define NBUF   2
#define LOOKA  1
#define WAIT_LANDED 32           // 1 chunk x 32 b32-instructions
#endif

#if defined(__HIP_DEVICE_COMPILE__) && __has_builtin(__builtin_amdgcn_s_wait_asynccnt)
#define WAIT_ASYNC(n) __builtin_amdgcn_s_wait_asynccnt(n)
#else
#define WAIT_ASYNC(n) asm volatile("s_wait_asynccnt %0" :: "i"(n) : "memory")
#endif

__device__ __forceinline__ float tanh_fast(float v) {
#if defined(__HIP_DEVICE_COMPILE__) && __has_builtin(__builtin_amdgcn_tanhf)
  return __builtin_amdgcn_tanhf(v);
#elif defined(__HIP_DEVICE_COMPILE__) && __has_builtin(__builtin_amdgcn_tanh_f32)
  return __builtin_amdgcn_tanh_f32(v);
#else
  return tanhf(v);
#endif
}

// clamp(v, 0.01, 1.0) as a single v_med3_f32 (keeps the recurrence chain at 2 deps/step)
__device__ __forceinline__ float clamp_r(float v) {
#if defined(__HIP_DEVICE_COMPILE__) && __has_builtin(__builtin_amdgcn_fmed3f)
  return __builtin_amdgcn_fmed3f(v, 0.01f, 1.0f);
#else
  return fminf(fmaxf(v, 0.01f), 1.0f);
#endif
}

__device__ __forceinline__ float sigmoid_f(float v) { return 1.0f / (1.0f + expf(-v)); }
__device__ __forceinline__ float softplus_f(float v) {
  return fmaxf(v, 0.0f) + log1pf(expf(-fabsf(v)));
}
__device__ __forceinline__ float gelu_f(float xv) {
  float x3 = xv * xv * xv;
  float t  = tanh_fast(0.7978845608028654f * fmaf(0.044715f, x3, xv));
  float h  = 0.5f * xv;
  return fmaf(h, t, h);
}

// MODE 0: per-(b,t) partial sums of r_combined (one wave owns 32 d-channels).
// MODE 1: recompute scan, write out = gelu(x) * r_combined * inv[b][t] (NT stores).
template <int MODE>
__global__ __launch_bounds__(DT)
void k_scan(const float* __restrict__ x,
            const float* __restrict__ pUf, const float* __restrict__ pTf,
            const float* __restrict__ pUs, const float* __restrict__ pTs,
            const float* __restrict__ pWf, const float* __restrict__ pWs,
            const float* __restrict__ emaf, const float* __restrict__ emas,
            const float* __restrict__ inv, float* __restrict__ out,
            float* __restrict__ psums)
{
#if HAVE_ASYNC
  __shared__ float tile[NBUF][CH * DT];   // staged x chunks
  __shared__ float rcbuf[CH * 33];        // padded transpose buffer (MODE 0)
#endif
  const int lane = threadIdx.x;
  const int b    = blockIdx.x / NBLK;
  const int blk  = blockIdx.x % NBLK;
  const int d    = blk * DT + lane;

  // Detached scalar parameters (derived once per thread; negligible).
  const float Uf   = sigmoid_f(pUf[0]);
  const float Us   = sigmoid_f(pUs[0]);
  const float tauf = fmaxf(softplus_f(pTf[0]), 0.5f);
  const float taus = fmaxf(softplus_f(pTs[0]), 0.5f);
  const float recf = 1.0f - expf(-1.0f / tauf);
  const float recs = 1.0f - expf(-1.0f / taus);
  const float wf   = sigmoid_f(pWf[0]);
  const float ws   = sigmoid_f(pWs[0]);
  const float mf   = 1.0f - recf;      // r' = clamp(r*(mf - a_f) + recf)
  const float ms   = 1.0f - recs;

  float r_f = emaf[d];
  float r_s = emas[d];

  const float* xbase = x + (size_t)b * T_LEN * D_LEN + blk * DT;  // chunk base (no lane)

#if HAVE_ASYNC
#if HAVE_ASYNC128
  // One b128 per 8-lane group stages 4 timesteps: lane -> (row = 4k + lane/8, col = (lane%8)*4)
  const int rowoff = lane >> 3;
  const int coloff = (lane & 7) << 2;
  auto issue_chunk = [&](int c, int buf) {
    const float* g = xbase + (size_t)c * CH * D_LEN;
    float* l = &tile[buf][0];
    #pragma unroll
    for (int k = 0; k < 8; ++k) {
      int row = 4 * k + rowoff;
      __builtin_amdgcn_global_load_async_to_lds_b128(
          (GV4*)(g + (size_t)row * D_LEN + coloff),
          (LV4*)(l + row * DT + coloff), 0, 0);
    }
  };
#else
  auto issue_chunk = [&](int c, int buf) {
    const float* g = xbase + (size_t)c * CH * D_LEN + lane;
    float* l = &tile[buf][lane];
    #pragma unroll
    for (int s = 0; s < CH; ++s)
      __builtin_amdgcn_global_load_async_to_lds_b32(
          (GInt*)(g + (size_t)s * D_LEN),
          (LInt*)(l + s * DT), 0, 0);
  };
#endif
#endif

  auto process_chunk = [&](int c, int buf) {
    float  ivl  = 0.0f;
    float* orow = nullptr;
    if (MODE == 1) {
      ivl  = inv[(size_t)b * T_LEN + c * CH + lane];
      orow = out + ((size_t)b * T_LEN + (size_t)c * CH) * D_LEN + d;
    }
    #pragma unroll
    for (int s = 0; s < CH; ++s) {
#if HAVE_ASYNC
      float xv = tile[buf][s * DT + lane];
#else
      float xv = xbase[(size_t)(c * CH + s) * D_LEN + lane];
#endif
      float o  = gelu_f(xv);
      float fr = tanh_fast(fabsf(o));
      float rc = fmaf(wf, r_f, ws * r_s);   // trace uses r BEFORE update

      if (MODE == 0) {
#if HAVE_ASYNC
        rcbuf[s * 33 + lane] = rc;          // 33-pad: conflict-free rows & columns
#else
        float tot = rc;
        #pragma unroll
        for (int msk = 16; msk >= 1; msk >>= 1) tot += __shfl_xor(tot, msk, 32);
        if (lane == s) psums[(size_t)blockIdx.x * T_LEN + c * CH + s] = tot;
#endif
      } else {
        float iv = __shfl(ivl, s, 32);
        __builtin_nontemporal_store(o * rc * iv, orow + (size_t)s * D_LEN);
      }

      // used = r*min(U*f, 0.99)  ->  r' = clamp(fma(r, (1-rec)-a, rec), .01, 1)
      float af = fminf(Uf * fr, 0.99f);
      r_f = clamp_r(fmaf(r_f, mf - af, recf));
      float as_ = fminf(Us * fr, 0.99f);
      r_s = clamp_r(fmaf(r_s, ms - as_, recs));
    }
#if HAVE_ASYNC
    if (MODE == 0) {
      asm volatile("" ::: "memory");        // keep rcbuf stores before the reads
      float acc = 0.0f;                     // lane sums row 'lane' (= step t0+lane)
      #pragma unroll
      for (int j = 0; j < CH; ++j) acc += rcbuf[lane * 33 + j];
      psums[(size_t)blockIdx.x * T_LEN + c * CH + lane] = acc;
    }
#endif
  };

#if HAVE_ASYNC
  #pragma unroll
  for (int p = 0; p < LOOKA; ++p) issue_chunk(p, p % NBUF);   // prologue

  int ib = LOOKA % NBUF;    // next issue buffer
  int cb = 0;               // current compute buffer
  int c  = 0;
  for (; c < NCHUNK - LOOKA; ++c) {
    issue_chunk(c + LOOKA, ib);
    if (++ib == NBUF) ib = 0;
    WAIT_ASYNC(WAIT_LANDED);                // chunk c fully landed (in-order)
    asm volatile("" ::: "memory");
    process_chunk(c, cb);
    if (++cb == NBUF) cb = 0;
  }
  WAIT_ASYNC(0);                            // tail: everything already requested
  asm volatile("" ::: "memory");
  for (; c < NCHUNK; ++c) {
    process_chunk(c, cb);
    if (++cb == NBUF) cb = 0;
  }
#else
  for (int c = 0; c < NCHUNK; ++c) process_chunk(c, 0);
#endif
}

__global__ __launch_bounds__(256)
void k_inv(const float* __restrict__ psums, float* __restrict__ inv)
{
  int i = blockIdx.x * 256 + threadIdx.x;          // over B*T
  if (i >= B_LEN * T_LEN) return;
  int b = i / T_LEN, t = i - b * T_LEN;
  float s = 0.0f;
  #pragma unroll
  for (int k = 0; k < NBLK; ++k)
    s += psums[(size_t)(b * NBLK + k) * T_LEN + t];
  // 1/(mean + eps) = D / (sum + D*eps)
  inv[i] = (float)D_LEN / (s + (float)D_LEN * 1e-5f);
}

extern "C" void kernel_launch(void* const* d_in, const int* in_sizes, int n_in,
                              void* d_out, int out_size, void* d_ws, size_t ws_size,
                              hipStream_t stream)
{
  (void)in_sizes; (void)n_in; (void)out_size; (void)ws_size;
  const float* x    = (const float*)d_in[0];
  const float* pUf  = (const float*)d_in[1];
  const float* pTf  = (const float*)d_in[2];
  const float* pUs  = (const float*)d_in[3];
  const float* pTs  = (const float*)d_in[4];
  const float* pWf  = (const float*)d_in[5];
  const float* pWs  = (const float*)d_in[6];
  const float* emaf = (const float*)d_in[7];
  const float* emas = (const float*)d_in[8];

  float* psums = (float*)d_ws;                                   // B*NBLK*T floats (4 MB)
  float* inv   = psums + (size_t)B_LEN * NBLK * T_LEN;           // B*T floats (128 KB)
  float* out   = (float*)d_out;

  dim3 grid(B_LEN * NBLK);   // 256 workgroups x 32 threads (1 wave each)
  k_scan<0><<<grid, DT, 0, stream>>>(x, pUf, pTf, pUs, pTs, pWf, pWs,
                                     emaf, emas, nullptr, nullptr, psums);
  k_inv<<<(B_LEN * T_LEN + 255) / 256, 256, 0, stream>>>(psums, inv);
  k_scan<1><<<grid, DT, 0, stream>>>(x, pUf, pTf, pUs, pTs, pWf, pWs,
                                     emaf, emas, inv, out, nullptr);
}